// AttentiveFPConv_42399917146354
// MI455X (gfx1250) — compile-verified
//
#include <hip/hip_runtime.h>

typedef __attribute__((ext_vector_type(2))) float v2f;
typedef __attribute__((ext_vector_type(8))) float v8f;

#define D 128

// One wave computes a 16x16 f32 tile of A[m0:m0+16, :] @ B[:, n0:n0+16], K = 128,
// using V_WMMA_F32_16X16X4_F32 (exact fp32 path).
// A row-major with leading dim `lda`; B row-major [128][128].
// f32 16x16x4 WMMA layouts (ISA 7.12.2):
//   A (16x4): lane L, vgpr j -> A[L%16][(L/16)*2 + j]
//   B (4x16): lane L, vgpr j -> B[(L/16)*2 + j][L%16]
//   C/D      : lane L, vgpr v -> C[v + (L/16)*8][L%16]
__device__ __forceinline__ v8f gemm_tile_f32(const float* __restrict__ A, int lda,
                                             const float* __restrict__ B,
                                             int m0, int n0, v8f acc) {
  const int lane = threadIdx.x & 31;
  const int l16  = lane & 15;
  const int half = lane >> 4;            // 0 or 1
  const float* arow = A + (size_t)(m0 + l16) * lda + half * 2;
  const float* bcol = B + (n0 + l16) + (size_t)(half * 2) * D;
#pragma unroll
  for (int k0 = 0; k0 < D; k0 += 4) {
    v2f a = *(const v2f*)(arow + k0);    // A[m][k0+half*2 + {0,1}], contiguous
    v2f b;
    b.x = bcol[(size_t)k0 * D];          // B[k0+half*2+0][n]
    b.y = bcol[(size_t)k0 * D + D];      // B[k0+half*2+1][n]
    acc = __builtin_amdgcn_wmma_f32_16x16x4_f32(
        /*neg_a=*/false, a, /*neg_b=*/false, b,
        /*c_mod=*/(short)0, acc, /*reuse_a=*/false, /*reuse_b=*/false);
  }
  return acc;
}

// ---- kernel 1: zero the aggregation buffer (float4 grid-stride) ----
__global__ __launch_bounds__(256) void zero_kernel(float4* __restrict__ p, int n4) {
  int i = blockIdx.x * blockDim.x + threadIdx.x;
  int stride = gridDim.x * blockDim.x;
  float4 z = {0.f, 0.f, 0.f, 0.f};
  for (; i < n4; i += stride) p[i] = z;
}

// ---- kernel 2: M = x * sigmoid(x @ Wa + ba)   (per-node gated message) ----
// Block = 256 threads = 8 waves; block handles 16 rows, wave w handles cols [16w,16w+16).
__global__ __launch_bounds__(256) void mgate_kernel(const float* __restrict__ x,
                                                    const float* __restrict__ Wa,
                                                    const float* __restrict__ ba,
                                                    float* __restrict__ M) {
  const int m0   = blockIdx.x * 16;
  const int n0   = (threadIdx.x >> 5) * 16;
  const int lane = threadIdx.x & 31;
  const int l16  = lane & 15;
  const int half = lane >> 4;

  v8f acc = {};
  acc = gemm_tile_f32(x, D, Wa, m0, n0, acc);

  const int   col  = n0 + l16;
  const float bias = ba[col];
#pragma unroll
  for (int v = 0; v < 8; ++v) {
    const int row = m0 + v + half * 8;
    const float t = acc[v] + bias;
    const float s = 1.0f / (1.0f + __expf(-t));       // sigmoid
    M[(size_t)row * D + col] = x[(size_t)row * D + col] * s;
  }
}

// ---- kernel 3: aggr[row[e]] += M[col[e]]  (one wave per edge) ----
// Lane handles 4 consecutive floats: float4 gather + 4 native f32 atomic adds.
// M and aggr are L2-resident (20.5 MB each << 192 MB L2).
__global__ __launch_bounds__(256) void scatter_kernel(const float* __restrict__ M,
                                                      const long long* __restrict__ ei,
                                                      float* __restrict__ aggr, int E) {
  const int wid  = (blockIdx.x * blockDim.x + threadIdx.x) >> 5;
  const int lane = threadIdx.x & 31;
  if (wid >= E) return;
  const int r = (int)ei[wid];        // edge_index[0][e] = destination row
  const int c = (int)ei[(size_t)E + wid];  // edge_index[1][e] = source col
  const float4 v = ((const float4*)(M + (size_t)c * D))[lane];
  float* dst = aggr + (size_t)r * D + lane * 4;
  __hip_atomic_fetch_add(dst + 0, v.x, __ATOMIC_RELAXED, __HIP_MEMORY_SCOPE_AGENT);
  __hip_atomic_fetch_add(dst + 1, v.y, __ATOMIC_RELAXED, __HIP_MEMORY_SCOPE_AGENT);
  __hip_atomic_fetch_add(dst + 2, v.z, __ATOMIC_RELAXED, __HIP_MEMORY_SCOPE_AGENT);
  __hip_atomic_fetch_add(dst + 3, v.w, __ATOMIC_RELAXED, __HIP_MEMORY_SCOPE_AGENT);
}

// ---- kernel 4: out = tanh(x @ Wn + bn + aggr @ Wg + bg) ----
// Two GEMMs fused into one accumulator: 64 WMMAs per wave-tile.
__global__ __launch_bounds__(256) void out_kernel(const float* __restrict__ x,
                                                  const float* __restrict__ aggr,
                                                  const float* __restrict__ Wn,
                                                  const float* __restrict__ bn,
                                                  const float* __restrict__ Wg,
                                                  const float* __restrict__ bg,
                                                  float* __restrict__ out) {
  const int m0   = blockIdx.x * 16;
  const int n0   = (threadIdx.x >> 5) * 16;
  const int lane = threadIdx.x & 31;
  const int l16  = lane & 15;
  const int half = lane >> 4;

  v8f acc = {};
  acc = gemm_tile_f32(x,    D, Wn, m0, n0, acc);
  acc = gemm_tile_f32(aggr, D, Wg, m0, n0, acc);

  const int   col  = n0 + l16;
  const float bias = bn[col] + bg[col];
#pragma unroll
  for (int v = 0; v < 8; ++v) {
    const int row = m0 + v + half * 8;
    out[(size_t)row * D + col] = tanhf(acc[v] + bias);
  }
}

extern "C" void kernel_launch(void* const* d_in, const int* in_sizes, int n_in,
                              void* d_out, int out_size, void* d_ws, size_t ws_size,
                              hipStream_t stream) {
  const float*     x  = (const float*)d_in[0];
  const long long* ei = (const long long*)d_in[1];   // int64 [2, E]
  const float*     Wn = (const float*)d_in[2];
  const float*     bn = (const float*)d_in[3];
  const float*     Wg = (const float*)d_in[4];
  const float*     bg = (const float*)d_in[5];
  const float*     Wa = (const float*)d_in[6];
  const float*     ba = (const float*)d_in[7];
  float*           out = (float*)d_out;

  const int N = in_sizes[0] / D;     // 40000 (divisible by 16)
  const int E = in_sizes[1] / 2;     // 640000

  float* Mbuf = (float*)d_ws;                       // N*D floats
  float* aggr = Mbuf + (size_t)N * D;               // N*D floats

  // 1) aggr = 0
  {
    int n4 = (N * D) / 4;
    int blocks = (n4 + 255) / 256;
    zero_kernel<<<blocks, 256, 0, stream>>>((float4*)aggr, n4);
  }
  // 2) M = x * sigmoid(x @ Wa + ba)
  mgate_kernel<<<N / 16, 256, 0, stream>>>(x, Wa, ba, Mbuf);
  // 3) aggr[row] += M[col]  over all edges
  {
    long long threads = (long long)E * 32;
    int blocks = (int)((threads + 255) / 256);
    scatter_kernel<<<blocks, 256, 0, stream>>>(Mbuf, ei, aggr, E);
  }
  // 4) out = tanh(x @ Wn + bn + aggr @ Wg + bg)
  out_kernel<<<N / 16, 256, 0, stream>>>(x, aggr, Wn, bn, Wg, bg, out);
}